// InstanceHead_81252191306378
// MI455X (gfx1250) — compile-verified
//
#include <hip/hip_runtime.h>
#include <hip/hip_bf16.h>
#include <stdint.h>

typedef __attribute__((ext_vector_type(2))) float v2f;
typedef __attribute__((ext_vector_type(8))) float v8f;
typedef __attribute__((ext_vector_type(4))) unsigned int v4u;
typedef __attribute__((ext_vector_type(4))) int v4i;
typedef __attribute__((ext_vector_type(8))) int v8i;

#define TAUV 0.1f
#define BB 4
#define SSZ 256
#define KK 128
#define LL 64
#define DD 32
#define HBITS 21
#define HSIZE (1u << HBITS)
#define HMASK (HSIZE - 1u)
#define NBINS 4096
#define CAND_CAP 2048
#define EMPTY64 0xFFFFFFFFFFFFFFFFull
#define TILE_ROWS 128
#define APAD 65  /* padded LDS row stride in DWORDs (TDM inserts 1 DWORD per 64) */

__device__ __forceinline__ uint32_t hashk(uint32_t k) {
    return (k * 2654435761u) & HMASK;
}

__device__ __forceinline__ int htab_lookup(const unsigned long long* table, uint32_t key) {
    uint32_t h = hashk(key);
    for (;;) {
        unsigned long long cur = table[h];
        if (cur == EMPTY64) return -1;
        if ((uint32_t)(cur >> 32) == key) return (int)(uint32_t)cur;
        h = (h + 1u) & HMASK;
    }
}

// ---------------- K1: clear hash table + histograms + counters ----------------
__global__ void k_clear(unsigned long long* table, int* hist, int* cnt) {
    int i = blockIdx.x * blockDim.x + threadIdx.x;
    if (i < (int)HSIZE) table[i] = EMPTY64;
    if (i < BB * NBINS) hist[i] = 0;
    if (i < BB) cnt[i] = 0;
}

// ---------------- K2: insert points; min original index per key ----------------
__global__ void k_insert(const int* __restrict__ coords, int N, unsigned long long* table) {
    int i = blockIdx.x * blockDim.x + threadIdx.x;
    if (i >= N) return;
    uint32_t b = (uint32_t)coords[i * 4 + 0];
    uint32_t x = (uint32_t)coords[i * 4 + 1];
    uint32_t y = (uint32_t)coords[i * 4 + 2];
    uint32_t z = (uint32_t)coords[i * 4 + 3];
    uint32_t key = ((b * SSZ + x) * SSZ + y) * SSZ + z;
    unsigned long long val = ((unsigned long long)key << 32) | (uint32_t)i;
    uint32_t h = hashk(key);
    for (;;) {
        unsigned long long cur = table[h];
        if (cur == EMPTY64) {
            unsigned long long prev = atomicCAS(&table[h], EMPTY64, val);
            if (prev == EMPTY64) return;
            cur = prev;
        }
        if ((uint32_t)(cur >> 32) == key) {
            atomicMin(&table[h], val);
            return;
        }
        h = (h + 1u) & HMASK;
    }
}

// ---------------- K3: neighborhood hmax, peak flag, conf histogram ----------------
__global__ void k_peak(const int* __restrict__ coords, const float* __restrict__ conf, int N,
                       const unsigned long long* __restrict__ table,
                       unsigned char* __restrict__ flag, int* __restrict__ hist) {
    int i = blockIdx.x * blockDim.x + threadIdx.x;
    if (i >= N) return;
    int b = coords[i * 4 + 0];
    int x = coords[i * 4 + 1];
    int y = coords[i * 4 + 2];
    int z = coords[i * 4 + 3];
    float ci = conf[i];
    bool mi = ci > TAUV;
    float hmax = -INFINITY;
    #pragma unroll
    for (int dx = -1; dx <= 1; ++dx)
        for (int dy = -1; dy <= 1; ++dy)
            for (int dz = -1; dz <= 1; ++dz) {
                int nx = x + dx, ny = y + dy, nz = z + dz;
                if ((unsigned)nx < SSZ && (unsigned)ny < SSZ && (unsigned)nz < SSZ) {
                    uint32_t nkey = (((uint32_t)b * SSZ + (uint32_t)nx) * SSZ + (uint32_t)ny) * SSZ + (uint32_t)nz;
                    int j = htab_lookup(table, nkey);
                    if (j >= 0) {
                        float cj = conf[j];
                        if (cj > TAUV) hmax = fmaxf(hmax, cj);
                    }
                }
            }
    bool peak = mi && (hmax == ci);
    flag[i] = peak ? 1 : 0;
    if (peak) {
        int bin = (int)(ci * (float)NBINS);
        bin = min(NBINS - 1, max(0, bin));
        atomicAdd(&hist[b * NBINS + bin], 1);
    }
}

// ---------------- K4: per-batch threshold bin ----------------
__global__ void k_thresh(const int* __restrict__ hist, int* __restrict__ thr) {
    int b = threadIdx.x;
    if (b >= BB) return;
    int cum = 0, t = 0;
    for (int bin = NBINS - 1; bin >= 0; --bin) {
        cum += hist[b * NBINS + bin];
        if (cum >= KK) { t = bin; break; }
    }
    thr[b] = t;
}

// ---------------- K5: collect candidates above threshold ----------------
__global__ void k_collect(const int* __restrict__ coords, const float* __restrict__ conf,
                          const unsigned char* __restrict__ flag, int N,
                          const int* __restrict__ thr, int* __restrict__ cnt,
                          unsigned long long* __restrict__ cand) {
    int i = blockIdx.x * blockDim.x + threadIdx.x;
    if (i >= N) return;
    if (!flag[i]) return;
    int b = coords[i * 4 + 0];
    float ci = conf[i];
    int bin = (int)(ci * (float)NBINS);
    bin = min(NBINS - 1, max(0, bin));
    if (bin < thr[b]) return;
    int slot = atomicAdd(&cnt[b], 1);
    if (slot < CAND_CAP) {
        unsigned long long k = ((unsigned long long)(~__float_as_uint(ci)) << 32) | (uint32_t)i;
        cand[b * CAND_CAP + slot] = k;
    }
}

// ---------------- K6: per-batch bitonic top-K ----------------
__global__ void k_topk(const unsigned long long* __restrict__ cand, const int* __restrict__ cnt,
                       float* __restrict__ peak_conf, int* __restrict__ selIdx) {
    __shared__ unsigned long long keys[CAND_CAP];
    int b = blockIdx.x;
    int count = min(cnt[b], CAND_CAP);
    for (int i = threadIdx.x; i < CAND_CAP; i += blockDim.x)
        keys[i] = (i < count) ? cand[b * CAND_CAP + i] : EMPTY64;
    __syncthreads();
    for (unsigned k = 2; k <= CAND_CAP; k <<= 1) {
        for (unsigned j = k >> 1; j > 0; j >>= 1) {
            for (unsigned i = threadIdx.x; i < CAND_CAP; i += blockDim.x) {
                unsigned ixj = i ^ j;
                if (ixj > i) {
                    unsigned long long a = keys[i];
                    unsigned long long c = keys[ixj];
                    bool up = ((i & k) == 0);
                    if ((up && a > c) || (!up && a < c)) { keys[i] = c; keys[ixj] = a; }
                }
            }
            __syncthreads();
        }
    }
    if (threadIdx.x < KK) {
        unsigned long long e = keys[threadIdx.x];
        bool valid = (threadIdx.x < count);
        float c = __uint_as_float(~(uint32_t)(e >> 32));
        int idx = (int)(uint32_t)e;
        peak_conf[b * KK + threadIdx.x] = valid ? c : 0.0f;
        selIdx[b * KK + threadIdx.x] = valid ? idx : -1;
    }
}

// ---------------- K7: neighbor feature average for 512 selected rows ----------------
__global__ void k_favg(const int* __restrict__ coords, const float* __restrict__ feats,
                       const float* __restrict__ conf, const unsigned long long* __restrict__ table,
                       const int* __restrict__ selIdx, float* __restrict__ cfeat) {
    int r = blockIdx.x;   // 0..B*K-1
    int t = threadIdx.x;  // 0..63
    int sel = selIdx[r];
    if (sel < 0) { cfeat[r * LL + t] = 0.0f; return; }
    int b = coords[sel * 4 + 0];
    int x = coords[sel * 4 + 1];
    int y = coords[sel * 4 + 2];
    int z = coords[sel * 4 + 3];
    float fsum = 0.0f, cntv = 0.0f;
    for (int dx = -1; dx <= 1; ++dx)
        for (int dy = -1; dy <= 1; ++dy)
            for (int dz = -1; dz <= 1; ++dz) {
                int nx = x + dx, ny = y + dy, nz = z + dz;
                if ((unsigned)nx < SSZ && (unsigned)ny < SSZ && (unsigned)nz < SSZ) {
                    uint32_t nkey = (((uint32_t)b * SSZ + (uint32_t)nx) * SSZ + (uint32_t)ny) * SSZ + (uint32_t)nz;
                    int j = htab_lookup(table, nkey);
                    if (j >= 0 && conf[j] > TAUV) {
                        fsum += feats[(long)j * LL + t];
                        cntv += 1.0f;
                    }
                }
            }
    cfeat[r * LL + t] = fsum / fmaxf(cntv, 1.0f);
}

// -------- TDM: issue a 1-D tensor_load_to_lds of one 128x64 f32 tile (32 KB),
// -------- padded 1 DWORD per 64 DWORDs so LDS row stride is 65 DWORDs.
__device__ __forceinline__ void tdm_load_tile(const float* feats, long totalElems,
                                              int tile, float* ldsDst) {
    uint64_t ga = (uint64_t)(uintptr_t)feats + (uint64_t)tile * (TILE_ROWS * LL * 4);
    uint32_t lds = (uint32_t)(uintptr_t)ldsDst;  // LDS byte offset (addr[31:0])
    long remain = totalElems - (long)tile * (TILE_ROWS * LL);
    if (remain < 0) remain = 0;
    uint32_t tdim0 = (uint32_t)remain;  // OOB elements read as zero

    v4u g0;
    g0.x = 1u;                                        // count=1, user descriptor
    g0.y = lds;                                       // lds_addr
    g0.z = (uint32_t)ga;                              // global_addr[31:0]
    g0.w = (uint32_t)(ga >> 32) | (2u << 30);         // global_addr[56:32] | type=2

    v8i g1;
    g1[0] = (int)((2u << 16) | (1u << 20) | (5u << 22)); // data_size=4B, pad_en, pad_interval=64dw, pad=1dw
    g1[1] = (int)((tdim0 & 0xFFFFu) << 16);              // tensor_dim0[15:0]
    g1[2] = (int)((tdim0 >> 16) | (1u << 16));           // tensor_dim0[31:16], tensor_dim1=1
    g1[3] = (int)((uint32_t)(TILE_ROWS * LL) << 16);     // tile_dim0 = 8192
    g1[4] = 0;                                           // tile_dim1=0, tile_dim2=0
    g1[5] = LL;                                          // tensor_dim0_stride (unused, 1-D)
    g1[6] = 0;
    g1[7] = 0;

    v4i z4 = {0, 0, 0, 0};
#if __clang_major__ >= 23
    v8i z8 = {0, 0, 0, 0, 0, 0, 0, 0};
    __builtin_amdgcn_tensor_load_to_lds(g0, g1, z4, z4, z8, 0);
#else
    __builtin_amdgcn_tensor_load_to_lds(g0, g1, z4, z4, 0);
#endif
}

// ---------------- K8: voxel_desc = l2norm(feats @ W + b) ----------------
// TDM-staged A tiles (double buffered) + WMMA f32 16x16x4.
__global__ __launch_bounds__(256) void k_desc(const float* __restrict__ feats,
                                              const float* __restrict__ W,
                                              const float* __restrict__ bias,
                                              float* __restrict__ out, int N,
                                              int ntiles, int tilesPerBlock) {
    __shared__ float Ws[LL * DD];
    __shared__ float Bs[DD];
    __shared__ float Abuf[2][TILE_ROWS * APAD];

    for (int i = threadIdx.x; i < LL * DD; i += blockDim.x) Ws[i] = W[i];
    if (threadIdx.x < DD) Bs[threadIdx.x] = bias[threadIdx.x];

    int t0 = blockIdx.x * tilesPerBlock;
    int t1 = min(ntiles, t0 + tilesPerBlock);
    if (t0 >= ntiles) return;

    int wave = threadIdx.x >> 5;
    int lane = threadIdx.x & 31;
    int m = lane & 15;     // A row in tile / B column
    int half = lane >> 4;  // k sub-group
    long totalElems = (long)N * LL;

    if (wave == 0) tdm_load_tile(feats, totalElems, t0, &Abuf[0][0]);

    for (int t = t0; t < t1; ++t) {
        int buf = (t - t0) & 1;
        if (wave == 0) {
            if (t + 1 < t1) {
                tdm_load_tile(feats, totalElems, t + 1, &Abuf[buf ^ 1][0]);
                __builtin_amdgcn_s_wait_tensorcnt(1);
            } else {
                __builtin_amdgcn_s_wait_tensorcnt(0);
            }
        }
        __syncthreads();  // tile t resident; previous compute finished

        const float* A = &Abuf[buf][(wave * 16 + m) * APAD];
        v8f c0 = {0.f, 0.f, 0.f, 0.f, 0.f, 0.f, 0.f, 0.f};
        v8f c1 = {0.f, 0.f, 0.f, 0.f, 0.f, 0.f, 0.f, 0.f};
        #pragma unroll
        for (int k0 = 0; k0 < LL; k0 += 4) {
            int kk = k0 + half * 2;
            v2f a;
            a.x = A[kk];
            a.y = A[kk + 1];
            v2f b0, b1;
            b0.x = Ws[kk * DD + m];
            b0.y = Ws[(kk + 1) * DD + m];
            b1.x = Ws[kk * DD + m + 16];
            b1.y = Ws[(kk + 1) * DD + m + 16];
            c0 = __builtin_amdgcn_wmma_f32_16x16x4_f32(false, a, false, b0, (short)0, c0, false, false);
            c1 = __builtin_amdgcn_wmma_f32_16x16x4_f32(false, a, false, b1, (short)0, c1, false, false);
        }
        float bl = Bs[m], bh = Bs[m + 16];
        long rowBase = (long)t * TILE_ROWS + wave * 16;
        #pragma unroll
        for (int v = 0; v < 8; ++v) {
            float x0 = c0[v] + bl;
            float x1 = c1[v] + bh;
            float s = x0 * x0 + x1 * x1;
            s += __shfl_xor(s, 1);
            s += __shfl_xor(s, 2);
            s += __shfl_xor(s, 4);
            s += __shfl_xor(s, 8);
            float invn = 1.0f / fmaxf(sqrtf(s), 1e-12f);
            long row = rowBase + v + half * 8;
            if (row < N) {
                out[row * DD + m] = x0 * invn;
                out[row * DD + m + 16] = x1 * invn;
            }
        }
        __syncthreads();  // all waves done reading buf before it is re-filled
    }
}

// ---------------- K9: full_desc (background + l2norm(cfeat @ W + b)) ----------------
__global__ __launch_bounds__(256) void k_cdesc(const float* __restrict__ cfeat,
                                               const float* __restrict__ W,
                                               const float* __restrict__ bias,
                                               const float* __restrict__ bg,
                                               const int* __restrict__ selIdx,
                                               float* __restrict__ outFull) {
    __shared__ float Ws[LL * DD];
    __shared__ float Bs[DD];
    for (int i = threadIdx.x; i < LL * DD; i += blockDim.x) Ws[i] = W[i];
    if (threadIdx.x < DD) Bs[threadIdx.x] = bias[threadIdx.x];
    __syncthreads();

    int wave = threadIdx.x >> 5;
    int lane = threadIdx.x & 31;
    int m = lane & 15;
    int half = lane >> 4;
    int rowBase = blockIdx.x * 128 + wave * 16;  // 0..511 (4 blocks)
    int r = rowBase + m;

    v8f c0 = {0.f, 0.f, 0.f, 0.f, 0.f, 0.f, 0.f, 0.f};
    v8f c1 = {0.f, 0.f, 0.f, 0.f, 0.f, 0.f, 0.f, 0.f};
    #pragma unroll
    for (int k0 = 0; k0 < LL; k0 += 4) {
        int kk = k0 + half * 2;
        v2f a;
        a.x = cfeat[r * LL + kk];
        a.y = cfeat[r * LL + kk + 1];
        v2f b0, b1;
        b0.x = Ws[kk * DD + m];
        b0.y = Ws[(kk + 1) * DD + m];
        b1.x = Ws[kk * DD + m + 16];
        b1.y = Ws[(kk + 1) * DD + m + 16];
        c0 = __builtin_amdgcn_wmma_f32_16x16x4_f32(false, a, false, b0, (short)0, c0, false, false);
        c1 = __builtin_amdgcn_wmma_f32_16x16x4_f32(false, a, false, b1, (short)0, c1, false, false);
    }
    float bl = Bs[m], bh = Bs[m + 16];
    #pragma unroll
    for (int v = 0; v < 8; ++v) {
        float x0 = c0[v] + bl;
        float x1 = c1[v] + bh;
        float s = x0 * x0 + x1 * x1;
        s += __shfl_xor(s, 1);
        s += __shfl_xor(s, 2);
        s += __shfl_xor(s, 4);
        s += __shfl_xor(s, 8);
        float invn = 1.0f / fmaxf(sqrtf(s), 1e-12f);
        int row = rowBase + v + half * 8;
        bool valid = selIdx[row] >= 0;
        int outRow = (row >> 7) * (KK + 1) + 1 + (row & (KK - 1));
        outFull[outRow * DD + m] = valid ? x0 * invn : 0.0f;
        outFull[outRow * DD + m + 16] = valid ? x1 * invn : 0.0f;
    }
    // background rows
    if (blockIdx.x == 0 && threadIdx.x < BB * DD) {
        int bi = threadIdx.x / DD;
        int c = threadIdx.x % DD;
        outFull[(bi * (KK + 1)) * DD + c] = bg[c];
    }
}

extern "C" void kernel_launch(void* const* d_in, const int* in_sizes, int n_in,
                              void* d_out, int out_size, void* d_ws, size_t ws_size,
                              hipStream_t stream) {
    (void)n_in; (void)out_size; (void)ws_size;
    const int* coords = (const int*)d_in[0];
    const float* feats = (const float*)d_in[1];
    const float* conf = (const float*)d_in[2];
    const float* W = (const float*)d_in[3];
    const float* bias = (const float*)d_in[4];
    const float* bg = (const float*)d_in[5];
    int N = in_sizes[0] / 4;

    float* out = (float*)d_out;
    float* peak_conf = out;                       // B*K
    float* voxel_desc = out + BB * KK;            // N*D
    float* full_desc = voxel_desc + (long)N * DD; // B*(K+1)*D

    // workspace layout
    char* w = (char*)d_ws;
    size_t off = 0;
    unsigned long long* table = (unsigned long long*)(w + off); off += (size_t)HSIZE * 8;
    int* hist = (int*)(w + off); off += (size_t)BB * NBINS * 4;
    int* thr = (int*)(w + off); off += 16;
    int* cnt = (int*)(w + off); off += 16;
    unsigned long long* cand = (unsigned long long*)(w + off); off += (size_t)BB * CAND_CAP * 8;
    int* selIdx = (int*)(w + off); off += (size_t)BB * KK * 4;
    float* cfeat = (float*)(w + off); off += (size_t)BB * KK * LL * 4;
    unsigned char* flag = (unsigned char*)(w + off); off += (size_t)N;

    int tb = 256;
    k_clear<<<(HSIZE + tb - 1) / tb, tb, 0, stream>>>(table, hist, cnt);
    k_insert<<<(N + tb - 1) / tb, tb, 0, stream>>>(coords, N, table);
    k_peak<<<(N + tb - 1) / tb, tb, 0, stream>>>(coords, conf, N, table, flag, hist);
    k_thresh<<<1, 32, 0, stream>>>(hist, thr);
    k_collect<<<(N + tb - 1) / tb, tb, 0, stream>>>(coords, conf, flag, N, thr, cnt, cand);
    k_topk<<<BB, 256, 0, stream>>>(cand, cnt, peak_conf, selIdx);
    k_favg<<<BB * KK, 64, 0, stream>>>(coords, feats, conf, table, selIdx, cfeat);

    int ntiles = (N + TILE_ROWS - 1) / TILE_ROWS;
    int nblk = ntiles < 625 ? ntiles : 625;
    int tilesPerBlock = (ntiles + nblk - 1) / nblk;
    k_desc<<<nblk, 256, 0, stream>>>(feats, W, bias, voxel_desc, N, ntiles, tilesPerBlock);
    k_cdesc<<<BB * KK / 128, 256, 0, stream>>>(cfeat, W, bias, bg, selIdx, full_desc);
}